// RiemannianGroupNorm_57784490000604
// MI455X (gfx1250) — compile-verified
//
#include <hip/hip_runtime.h>
#include <math.h>

// ---------------- constants mirroring the reference ----------------
#define EPSF      1e-7f
#define ACLIP     (1.0f - 1e-5f)
#define DIM       64
#define NBLK      512
#define WPB       8              // waves per block (256 threads, wave32)
#define N_ITERS   10

typedef __attribute__((ext_vector_type(2))) float v2f;
typedef __attribute__((ext_vector_type(8))) float v8f;

struct F2 { float x, y; };

// wave32 butterfly reduction (deterministic)
__device__ inline float wred(float v) {
#pragma unroll
  for (int o = 16; o > 0; o >>= 1) v += __shfl_xor(v, o, 32);
  return v;
}
__device__ inline float dotred(F2 a, F2 b) { return wred(a.x * b.x + a.y * b.y); }

// mobius_add(X, Y) given dots; per-lane components, scalar coefficients
__device__ inline F2 mob(F2 X, F2 Y, float xy, float x2, float y2) {
  float alpha = 1.f + 2.f * xy + y2;
  float beta  = 1.f - x2;
  float den   = fmaxf(1.f + 2.f * xy + x2 * y2, EPSF);
  F2 r;
  r.x = (alpha * X.x + beta * Y.x) / den;
  r.y = (alpha * X.y + beta * Y.y) / den;
  return r;
}

// ---------------- tiny: q = exp0(mean) ----------------
__global__ void k_init_q(const float* __restrict__ mean, float* __restrict__ q) {
  int l = threadIdx.x;                      // 32 lanes, comps (2l, 2l+1)
  F2 v; v.x = mean[2 * l]; v.y = mean[2 * l + 1];
  float nn  = dotred(v, v);
  float nrm = sqrtf(fmaxf(nn, EPSF));
  float c   = tanhf(nrm) / nrm;
  q[2 * l] = c * v.x; q[2 * l + 1] = c * v.y;
}

// ---------------- partial sums of log0(x_i) ----------------
__global__ void k_reduce_log0(const float* __restrict__ x, float* __restrict__ vp, int n) {
  __shared__ float lds[WPB * DIM];
  int lane = threadIdx.x & 31, wv = threadIdx.x >> 5;
  int gw = blockIdx.x * WPB + wv, tw = gridDim.x * WPB;
  F2 acc; acc.x = 0.f; acc.y = 0.f;
  for (int r = gw; r < n; r += tw) {
    const float* row = x + (size_t)r * DIM;
    F2 xv; xv.x = row[2 * lane]; xv.y = row[2 * lane + 1];
    float xx  = dotred(xv, xv);
    float nrm = sqrtf(fmaxf(xx, EPSF));
    float c   = atanhf(fminf(nrm, ACLIP)) / nrm;
    acc.x += c * xv.x; acc.y += c * xv.y;
  }
  lds[wv * DIM + 2 * lane] = acc.x; lds[wv * DIM + 2 * lane + 1] = acc.y;
  __syncthreads();
  if (threadIdx.x < DIM) {
    float s = 0.f;
#pragma unroll
    for (int w = 0; w < WPB; ++w) s += lds[w * DIM + threadIdx.x];
    vp[blockIdx.x * DIM + threadIdx.x] = s;
  }
}

// ---------------- partial sums of logmap(mu, x_i) ----------------
__global__ void k_reduce_logmap(const float* __restrict__ x, const float* __restrict__ mu,
                                float* __restrict__ vp, int n) {
  __shared__ float lds[WPB * DIM];
  int lane = threadIdx.x & 31, wv = threadIdx.x >> 5;
  int gw = blockIdx.x * WPB + wv, tw = gridDim.x * WPB;
  F2 m; m.x = mu[2 * lane]; m.y = mu[2 * lane + 1];
  float pp = dotred(m, m);
  float two_over_lam = fmaxf(1.f - pp, EPSF);   // 2/lambda_p
  F2 a; a.x = -m.x; a.y = -m.y;                 // a = -mu
  F2 acc; acc.x = 0.f; acc.y = 0.f;
  for (int r = gw; r < n; r += tw) {
    const float* row = x + (size_t)r * DIM;
    F2 xv; xv.x = row[2 * lane]; xv.y = row[2 * lane + 1];
    float xx = dotred(xv, xv);
    float px = dotred(m, xv);
    float ab = -px;
    float alpha = 1.f + 2.f * ab + xx;
    float beta  = 1.f - pp;
    float den   = fmaxf(1.f + 2.f * ab + pp * xx, EPSF);
    // ||w||^2 computed algebraically in span{a, x} (saves a butterfly)
    float ww = fmaxf((alpha * alpha * pp + 2.f * alpha * beta * ab + beta * beta * xx)
                     / (den * den), EPSF);
    float nw   = sqrtf(ww);
    float coef = two_over_lam * atanhf(fminf(nw, ACLIP)) / nw / den;
    acc.x += coef * (alpha * a.x + beta * xv.x);
    acc.y += coef * (alpha * a.y + beta * xv.y);
  }
  lds[wv * DIM + 2 * lane] = acc.x; lds[wv * DIM + 2 * lane + 1] = acc.y;
  __syncthreads();
  if (threadIdx.x < DIM) {
    float s = 0.f;
#pragma unroll
    for (int w = 0; w < WPB; ++w) s += lds[w * DIM + threadIdx.x];
    vp[blockIdx.x * DIM + threadIdx.x] = s;
  }
}

// ---- stage-2: colsum of 512x64 partials via FP32 WMMA, then mu update ----
// mode 0: mu = exp0(sum/n)        (Frechet-mean init)
// mode 1: mu = expmap(mu, sum/n)  (Karcher step)
__global__ void k_finish_mu(const float* __restrict__ vp, float* __restrict__ mu,
                            int n, int mode) {
  __shared__ float s[DIM];
  int lane = threadIdx.x;               // 32 threads, one wave, EXEC all-ones
  int half = lane >> 4, li = lane & 15;
  v2f A = {1.f, 1.f};                   // ones(16x4): C[m,n] = sum_k B[k,n]
  v8f c0 = {}, c1 = {}, c2 = {}, c3 = {};
  for (int k0 = 0; k0 < NBLK; k0 += 4) {
    const float* r0 = vp + (size_t)(k0 + 2 * half) * DIM;   // rows k0+{0,2}
    const float* r1 = r0 + DIM;                             // rows k0+{1,3}
    v2f b;
    b[0] = r0[0 * 16 + li]; b[1] = r1[0 * 16 + li];
    c0 = __builtin_amdgcn_wmma_f32_16x16x4_f32(false, A, false, b, (short)0, c0, false, false);
    b[0] = r0[1 * 16 + li]; b[1] = r1[1 * 16 + li];
    c1 = __builtin_amdgcn_wmma_f32_16x16x4_f32(false, A, false, b, (short)0, c1, false, false);
    b[0] = r0[2 * 16 + li]; b[1] = r1[2 * 16 + li];
    c2 = __builtin_amdgcn_wmma_f32_16x16x4_f32(false, A, false, b, (short)0, c2, false, false);
    b[0] = r0[3 * 16 + li]; b[1] = r1[3 * 16 + li];
    c3 = __builtin_amdgcn_wmma_f32_16x16x4_f32(false, A, false, b, (short)0, c3, false, false);
  }
  // every C row is identical; lane l's element 0 holds colsum[(l&15)] of its group
  if (lane < 16) {
    s[ 0 + lane] = c0[0]; s[16 + lane] = c1[0];
    s[32 + lane] = c2[0]; s[48 + lane] = c3[0];
  }
  __syncthreads();
  float inv_n = 1.f / (float)n;
  F2 g; g.x = s[2 * lane] * inv_n; g.y = s[2 * lane + 1] * inv_n;
  if (mode == 0) {
    float nn = dotred(g, g);
    float nrm = sqrtf(fmaxf(nn, EPSF));
    float c = tanhf(nrm) / nrm;
    mu[2 * lane] = c * g.x; mu[2 * lane + 1] = c * g.y;
  } else {
    F2 p; p.x = mu[2 * lane]; p.y = mu[2 * lane + 1];
    float pp  = dotred(p, p);
    float nn  = dotred(g, g);
    float nrm = sqrtf(fmaxf(nn, EPSF));
    float lam = 2.f / fmaxf(1.f - pp, EPSF);
    float tc  = tanhf(0.5f * lam * nrm) / nrm;
    F2 t; t.x = tc * g.x; t.y = tc * g.y;
    float pt = dotred(p, t);
    float tt = tc * tc * nn;
    F2 o = mob(p, t, pt, pp, tt);
    mu[2 * lane] = o.x; mu[2 * lane + 1] = o.y;
  }
}

// ---------------- partial sums of dist(mu, x_i)^2 ----------------
__global__ void k_reduce_var(const float* __restrict__ x, const float* __restrict__ mu,
                             float* __restrict__ sp, int n) {
  __shared__ float lds[WPB];
  int lane = threadIdx.x & 31, wv = threadIdx.x >> 5;
  int gw = blockIdx.x * WPB + wv, tw = gridDim.x * WPB;
  F2 m; m.x = mu[2 * lane]; m.y = mu[2 * lane + 1];
  float pp = dotred(m, m);
  float acc = 0.f;
  for (int r = gw; r < n; r += tw) {
    const float* row = x + (size_t)r * DIM;
    F2 xv; xv.x = row[2 * lane]; xv.y = row[2 * lane + 1];
    float xx = dotred(xv, xv);
    float px = dotred(m, xv);
    float ab = -px;
    float alpha = 1.f + 2.f * ab + xx;
    float beta  = 1.f - pp;
    float den   = fmaxf(1.f + 2.f * ab + pp * xx, EPSF);
    float ww = fmaxf((alpha * alpha * pp + 2.f * alpha * beta * ab + beta * beta * xx)
                     / (den * den), EPSF);
    float d = 2.f * atanhf(fminf(sqrtf(ww), ACLIP));
    acc += d * d;
  }
  if (lane == 0) lds[wv] = acc;
  __syncthreads();
  if (threadIdx.x == 0) {
    float s = 0.f;
#pragma unroll
    for (int w = 0; w < WPB; ++w) s += lds[w];
    sp[blockIdx.x] = s;
  }
}

// gs = sqrt(var / (mean(d^2) + 1e-6))
__global__ void k_finish_var(const float* __restrict__ sp, const float* __restrict__ var,
                             float* __restrict__ gs, int n) {
  int lane = threadIdx.x;
  float s = 0.f;
  for (int i = lane; i < NBLK; i += 32) s += sp[i];
  s = wred(s);
  if (lane == 0) gs[0] = sqrtf(var[0] / (s / (float)n + 1e-6f));
}

// ------ final: out_i = expmap(q, gs * (lam_p/lam_q) * gyr(q,-mu, logmap(mu,x_i))) ------
__global__ void k_final(const float* __restrict__ x, const float* __restrict__ mu,
                        const float* __restrict__ qg, const float* __restrict__ gs,
                        float* __restrict__ out, int n) {
  int lane = threadIdx.x & 31, wv = threadIdx.x >> 5;
  int gw = blockIdx.x * WPB + wv, tw = gridDim.x * WPB;
  // wave-uniform precompute
  F2 m; m.x = mu[2 * lane]; m.y = mu[2 * lane + 1];
  F2 q; q.x = qg[2 * lane]; q.y = qg[2 * lane + 1];
  float gscale = gs[0];
  float pp = dotred(m, m);
  float qq = dotred(q, q);
  F2 b; b.x = -m.x; b.y = -m.y;                 // b = -mu  (bb = pp)
  float qb = dotred(q, b);
  F2 sv = mob(q, b, qb, qq, pp);                // s = mobius_add(q, -mu)
  float ss = dotred(sv, sv);
  F2 msv; msv.x = -sv.x; msv.y = -sv.y;
  float ratio = fmaxf(1.f - qq, EPSF) / fmaxf(1.f - pp, EPSF);  // lam_p / lam_q
  float lamq  = 2.f / fmaxf(1.f - qq, EPSF);
  float two_over_lam = fmaxf(1.f - pp, EPSF);
  float sc = ratio * gscale;

  for (int r = gw; r < n; r += tw) {
    const float* row = x + (size_t)r * DIM;
    F2 xv; xv.x = row[2 * lane]; xv.y = row[2 * lane + 1];
    // v = logmap(mu, x)
    float xx = dotred(xv, xv);
    float px = dotred(m, xv);
    float ab = -px;
    float alpha = 1.f + 2.f * ab + xx;
    float beta  = 1.f - pp;
    float den   = fmaxf(1.f + 2.f * ab + pp * xx, EPSF);
    float ww = fmaxf((alpha * alpha * pp + 2.f * alpha * beta * ab + beta * beta * xx)
                     / (den * den), EPSF);
    float nw   = sqrtf(ww);
    float coef = two_over_lam * atanhf(fminf(nw, ACLIP)) / nw;
    F2 v; v.x = coef * (alpha * b.x + beta * xv.x) / den;
          v.y = coef * (alpha * b.y + beta * xv.y) / den;
    float vv = coef * coef * ww;
    // gyr(q, -mu, v) = mobius_add(-s, mobius_add(q, mobius_add(b, v)))
    float bv = dotred(b, v);
    F2 t1 = mob(b, v, bv, pp, vv);
    float t1t1 = dotred(t1, t1);
    float qt1  = dotred(q, t1);
    F2 t2 = mob(q, t1, qt1, qq, t1t1);
    float t2t2 = dotred(t2, t2);
    float st2  = dotred(msv, t2);
    F2 rv = mob(msv, t2, st2, ss, t2t2);
    // scale, then expmap at q
    F2 u; u.x = sc * rv.x; u.y = sc * rv.y;
    float uu = dotred(u, u);
    float nu = sqrtf(fmaxf(uu, EPSF));
    float tc = tanhf(0.5f * lamq * nu) / nu;
    F2 t; t.x = tc * u.x; t.y = tc * u.y;
    float qt = dotred(q, t);
    float tt = tc * tc * uu;
    F2 o = mob(q, t, qt, qq, tt);
    float* orow = out + (size_t)r * DIM;
    orow[2 * lane] = o.x; orow[2 * lane + 1] = o.y;
  }
}

extern "C" void kernel_launch(void* const* d_in, const int* in_sizes, int n_in,
                              void* d_out, int out_size, void* d_ws, size_t ws_size,
                              hipStream_t stream) {
  const float* x    = (const float*)d_in[0];
  const float* mean = (const float*)d_in[1];
  const float* var  = (const float*)d_in[2];
  float* out = (float*)d_out;
  int n = in_sizes[0] / DIM;

  float* ws = (float*)d_ws;
  float* mu = ws;                 // [64]
  float* q  = ws + 64;            // [64]
  float* gs = ws + 128;           // [1]
  float* vp = ws + 192;           // [NBLK * 64] vector partials
  float* sp = vp + NBLK * DIM;    // [NBLK]      scalar partials

  k_init_q<<<1, 32, 0, stream>>>(mean, q);
  k_reduce_log0<<<NBLK, 256, 0, stream>>>(x, vp, n);
  k_finish_mu<<<1, 32, 0, stream>>>(vp, mu, n, 0);
  for (int it = 0; it < N_ITERS; ++it) {
    k_reduce_logmap<<<NBLK, 256, 0, stream>>>(x, mu, vp, n);
    k_finish_mu<<<1, 32, 0, stream>>>(vp, mu, n, 1);
  }
  k_reduce_var<<<NBLK, 256, 0, stream>>>(x, mu, sp, n);
  k_finish_var<<<1, 32, 0, stream>>>(sp, var, gs, n);
  k_final<<<NBLK, 256, 0, stream>>>(x, mu, q, gs, out, n);
}